// Feedback_Encoder_10995116277876
// MI455X (gfx1250) — compile-verified
//
#include <hip/hip_runtime.h>

#define NUM_USERS 200000
#define NUM_ITEMS 100000
#define DIM 64
#define K_TYPES 4
#define OUT_STRIDE (K_TYPES * DIM)   // 256

typedef float v2f __attribute__((ext_vector_type(2)));
typedef float v8f __attribute__((ext_vector_type(8)));

// ---------------------------------------------------------------------------
// Zero a buffer (float4 vectorized)
// ---------------------------------------------------------------------------
__global__ void zero_f4(float4* __restrict__ p, long n4) {
    long i = (long)blockIdx.x * blockDim.x + threadIdx.x;
    if (i < n4) {
        float4 z; z.x = 0.f; z.y = 0.f; z.z = 0.f; z.w = 0.f;
        p[i] = z;
    }
}

// ---------------------------------------------------------------------------
// COO SpMM scatter: Y[rows[e]+row_off, :] += vals[e] * X[cols[e], :]
// One wave (32 lanes) per edge; each lane handles 2 of the 64 dims.
// Gather is a coalesced 256B row read (hot in 192MB L2); scatter uses
// hardware global_atomic_add_f32 (coalesced 256B across the wave).
// ---------------------------------------------------------------------------
__global__ void spmm_scatter(const int*  __restrict__ rows,
                             const int*  __restrict__ cols,
                             const float* __restrict__ vals,
                             int nE,
                             const float* __restrict__ X,
                             float* __restrict__ Y,
                             int row_off) {
    long gid  = (long)blockIdx.x * blockDim.x + threadIdx.x;
    int  lane = (int)(gid & 31);
    long e    = gid >> 5;
    if (e >= nE) return;

    int   r = rows[e] + row_off;   // same address across wave -> 1 request
    int   c = cols[e];
    float v = vals[e];

    v2f x = *(const v2f*)(X + (long)c * DIM + lane * 2);
    float* yp = Y + (long)r * DIM + lane * 2;
    unsafeAtomicAdd(yp,     v * x.x);
    unsafeAtomicAdd(yp + 1, v * x.y);
}

// ---------------------------------------------------------------------------
// c[i] = a[i] + b[i] + c[i]  (layer-mean accumulator, float4 vectorized)
// ---------------------------------------------------------------------------
__global__ void combine3(const float4* __restrict__ a,
                         const float4* __restrict__ b,
                         float4* __restrict__ c, long n4) {
    long i = (long)blockIdx.x * blockDim.x + threadIdx.x;
    if (i < n4) {
        float4 A = a[i], B = b[i], C = c[i];
        C.x += A.x + B.x;
        C.y += A.y + B.y;
        C.z += A.z + B.z;
        C.w += A.w + B.w;
        c[i] = C;
    }
}

// ---------------------------------------------------------------------------
// out[r, col_off + n] = relu(scale * sum_k A[r,k] * W[k,n])
// A: [rows, 64] f32, W: [64, 64] f32, out row stride = 256.
// blockDim = 128 (4 waves); wave w computes the 16x16 tile at
// (row0 = 16*blockIdx.x, n0 = 16*w) via 16 chained V_WMMA_F32_16X16X4_F32.
// ---------------------------------------------------------------------------
__global__ void gemm_wmma_relu(const float* __restrict__ A,
                               const float* __restrict__ W,
                               float* __restrict__ out,
                               int col_off, float scale) {
    const int wid  = threadIdx.x >> 5;       // 0..3 -> N tile
    const int lane = threadIdx.x & 31;
    const int row0 = blockIdx.x * 16;
    const int n0   = wid * 16;
    const int m    = lane & 15;              // A row / B,N column within tile
    const int kb   = (lane < 16) ? 0 : 2;    // K sub-offset for this half-wave

    v8f c = {};                              // C starts at 0
    const float* arow = A + (long)(row0 + m) * DIM;

#pragma unroll
    for (int kk = 0; kk < DIM; kk += 4) {
        // A 16x4: lane m holds K = kk+kb, kk+kb+1 in vgpr0/1
        v2f a = *(const v2f*)(arow + kk + kb);
        // B 4x16: lane (N=m) holds rows K = kk+kb, kk+kb+1 in vgpr0/1
        v2f b;
        b.x = W[(kk + kb)     * DIM + n0 + m];
        b.y = W[(kk + kb + 1) * DIM + n0 + m];
        c = __builtin_amdgcn_wmma_f32_16x16x4_f32(
                /*neg_a=*/false, a, /*neg_b=*/false, b,
                /*c_mod=*/(short)0, c, /*reuse_a=*/false, /*reuse_b=*/false);
    }

    // D 16x16: vgpr v holds row M = v + (lane<16 ? 0 : 8), N = lane&15
    const int mb = (lane < 16) ? 0 : 8;
#pragma unroll
    for (int v = 0; v < 8; ++v) {
        float val = c[v] * scale;
        val = val > 0.f ? val : 0.f;
        out[(long)(row0 + v + mb) * OUT_STRIDE + col_off + n0 + m] = val;
    }
}

// ---------------------------------------------------------------------------
// Host orchestration
// ---------------------------------------------------------------------------
extern "C" void kernel_launch(void* const* d_in, const int* in_sizes, int n_in,
                              void* d_out, int out_size, void* d_ws, size_t ws_size,
                              hipStream_t stream) {
    const int*   u2u_rows  = (const int*)  d_in[0];
    const int*   u2u_cols  = (const int*)  d_in[1];
    const float* u2u_vals  = (const float*)d_in[2];
    const int*   u2i_rows0 = (const int*)  d_in[3];
    const int*   u2i_cols0 = (const int*)  d_in[4];
    const float* u2i_vals0 = (const float*)d_in[5];
    const int*   u2i_rows1 = (const int*)  d_in[6];
    const int*   u2i_cols1 = (const int*)  d_in[7];
    const float* u2i_vals1 = (const float*)d_in[8];
    const int*   i2u_rows0 = (const int*)  d_in[9];
    const int*   i2u_cols0 = (const int*)  d_in[10];
    const float* i2u_vals0 = (const float*)d_in[11];
    const int*   i2u_rows1 = (const int*)  d_in[12];
    const int*   i2u_cols1 = (const int*)  d_in[13];
    const float* i2u_vals1 = (const float*)d_in[14];
    const int*   i2i_rows  = (const int*)  d_in[15];
    const int*   i2i_cols  = (const int*)  d_in[16];
    const float* i2i_vals  = (const float*)d_in[17];
    const float* user_embs = (const float*)d_in[18];  // [4, U, 64]
    const float* item_embs = (const float*)d_in[19];  // [4, I, 64]
    const float* W_u       = (const float*)d_in[20];  // [4, 64, 64]
    const float* W_v       = (const float*)d_in[21];  // [4, 64, 64]

    const int E_UU = in_sizes[0];
    const int E_UI = in_sizes[3];
    const int E_II = in_sizes[15];

    const long UN = (long)NUM_USERS * DIM;   // 12.8M floats
    const long IN = (long)NUM_ITEMS * DIM;   //  6.4M floats

    float* ws  = (float*)d_ws;
    float* ue1 = ws;              // layer-1 users, later the user accumulator
    float* ue2 = ws + UN;         // layer-2 users
    float* ie1 = ws + 2 * UN;     // layer-1 items, later the item accumulator
    float* ie2 = ie1 + IN;        // layer-2 items

    float* out_users = (float*)d_out;
    float* out_items = out_users + (long)NUM_USERS * OUT_STRIDE;

    auto zero = [&](float* p, long n) {
        long n4 = n / 4;
        zero_f4<<<(unsigned)((n4 + 255) / 256), 256, 0, stream>>>((float4*)p, n4);
    };
    auto spmm = [&](const int* r, const int* cI, const float* v, int nE,
                    const float* X, float* Y, int off) {
        long t = (long)nE * 32;
        spmm_scatter<<<(unsigned)((t + 255) / 256), 256, 0, stream>>>(r, cI, v, nE, X, Y, off);
    };
    auto comb = [&](const float* a, const float* b, float* cP, long n) {
        long n4 = n / 4;
        combine3<<<(unsigned)((n4 + 255) / 256), 256, 0, stream>>>(
            (const float4*)a, (const float4*)b, (float4*)cP, n4);
    };

    const float inv = 1.0f / 3.0f;  // mean over layers 0,1,2

    for (int k = 0; k < K_TYPES; ++k) {
        const float* ue0 = user_embs + (long)k * UN;
        const float* ie0 = item_embs + (long)k * IN;

        // ---- layer 1: (ue0, ie0) -> (ue1, ie1) ----
        zero(ue1, UN);
        zero(ie1, IN);
        spmm(u2u_rows,  u2u_cols,  u2u_vals,  E_UU, ue0, ue1, 0);
        spmm(u2i_rows0, u2i_cols0, u2i_vals0, E_UI, ie0, ue1, 0);
        spmm(u2i_rows1, u2i_cols1, u2i_vals1, E_UI, ie0, ue1, NUM_USERS / 2);
        spmm(i2i_rows,  i2i_cols,  i2i_vals,  E_II, ie0, ie1, 0);
        spmm(i2u_rows0, i2u_cols0, i2u_vals0, E_UI, ue0, ie1, 0);
        spmm(i2u_rows1, i2u_cols1, i2u_vals1, E_UI, ue0, ie1, NUM_ITEMS / 2);

        // ---- layer 2: (ue1, ie1) -> (ue2, ie2) ----
        zero(ue2, UN);
        zero(ie2, IN);
        spmm(u2u_rows,  u2u_cols,  u2u_vals,  E_UU, ue1, ue2, 0);
        spmm(u2i_rows0, u2i_cols0, u2i_vals0, E_UI, ie1, ue2, 0);
        spmm(u2i_rows1, u2i_cols1, u2i_vals1, E_UI, ie1, ue2, NUM_USERS / 2);
        spmm(i2i_rows,  i2i_cols,  i2i_vals,  E_II, ie1, ie2, 0);
        spmm(i2u_rows0, i2u_cols0, i2u_vals0, E_UI, ue1, ie2, 0);
        spmm(i2u_rows1, i2u_cols1, i2u_vals1, E_UI, ue1, ie2, NUM_ITEMS / 2);

        // ---- accumulate layer mean numerator in place: ue1 += ue0 + ue2 ----
        comb(ue0, ue2, ue1, UN);
        comb(ie0, ie2, ie1, IN);

        // ---- fused (acc * inv) @ W + ReLU via V_WMMA_F32_16X16X4_F32 ----
        gemm_wmma_relu<<<NUM_USERS / 16, 128, 0, stream>>>(
            ue1, W_u + (long)k * DIM * DIM, out_users, k * DIM, inv);
        gemm_wmma_relu<<<NUM_ITEMS / 16, 128, 0, stream>>>(
            ie1, W_v + (long)k * DIM * DIM, out_items, k * DIM, inv);
    }
}